// AdaptiveWeightedKnowledgeDistillationLoss_24206435680616
// MI455X (gfx1250) — compile-verified
//
#include <hip/hip_runtime.h>
#include <float.h>

// T = 4.0 exactly => exp(u) = (exp(u/4))^4, one v_exp_f32 per element.
#define T_DIST 4.0f
#define INV_T  0.25f

typedef __attribute__((ext_vector_type(2))) float v2f;
typedef __attribute__((ext_vector_type(8))) float v8f;

__device__ __forceinline__ float wave_sum(float v) {
#pragma unroll
    for (int off = 16; off > 0; off >>= 1) v += __shfl_xor(v, off, 32);
    return v;
}
__device__ __forceinline__ float wave_max(float v) {
#pragma unroll
    for (int off = 16; off > 0; off >>= 1) v = fmaxf(v, __shfl_xor(v, off, 32));
    return v;
}

// Online-softmax accumulate for one (student, teacher) logit pair.
__device__ __forceinline__ void kd_accum(float sx, float tx,
                                         float& ms, float& mt,
                                         float& a0, float& a1,
                                         float& b0, float& b1, float& b2,
                                         float& b3, float& b4) {
    // ---- student: a0 = sum e^(s-ms), a1 = sum e^((s-ms)/T)
    if (sx > ms) {
        float dT  = (ms - sx) * INV_T;
        float edT = __expf(dT);
        float ed  = edT * edT; ed *= ed;      // e^(d) = (e^(d/4))^4
        a0 *= ed; a1 *= edT;
        ms = sx;
    }
    float esT = __expf((sx - ms) * INV_T);
    float es  = esT * esT; es *= es;
    a0 += es; a1 += esT;

    // ---- teacher: b0 = sum e^u, b1 = sum u e^u, b2 = sum e^(u/T),
    //               b3 = sum e^(u/T)*t, b4 = sum e^(u/T)*s   (u = t - mt)
    if (tx > mt) {
        float d   = mt - tx;                  // <= 0
        float edT = __expf(d * INV_T);
        float ed  = edT * edT; ed *= ed;
        b1 = ed * (b1 + d * b0);              // affine rescale of sum u e^u
        b0 *= ed; b2 *= edT; b3 *= edT; b4 *= edT;
        mt = tx;
    }
    float u  = tx - mt;
    float eT = __expf(u * INV_T);
    float e1 = eT * eT; e1 *= e1;
    b0 += e1; b1 += u * e1;
    b2 += eT; b3 += eT * tx; b4 += eT * sx;
}

__global__ __launch_bounds__(256)
void kd_row_kernel(const float* __restrict__ s_all,
                   const float* __restrict__ t_all,
                   const long long* __restrict__ targets,
                   float* __restrict__ row_loss, int C) {
    const int row  = blockIdx.x;
    const int tid  = threadIdx.x;
    const int lane = tid & 31;
    const int wave = tid >> 5;                // 8 waves

    const float* s = s_all + (size_t)row * (size_t)C;
    const float* t = t_all + (size_t)row * (size_t)C;

    __shared__ float sP[8][16];               // wave partials, zero-padded to 16 "rows"
    __shared__ float sMax[8];

    float s_tgt = 0.0f;
    if (tid == 0) s_tgt = s[(int)targets[row]];

    float ms = -FLT_MAX, mt = -FLT_MAX;
    float a0 = 0, a1 = 0, b0 = 0, b1 = 0, b2 = 0, b3 = 0, b4 = 0;

    // ---- single streaming pass, 128-bit coalesced loads ----
    const int Cvec = C >> 2;
    const float4* s4 = (const float4*)s;
    const float4* t4 = (const float4*)t;
    for (int j = tid; j < Cvec; j += 256) {
        float4 sv = s4[j];
        float4 tv = t4[j];
        float se[4] = {sv.x, sv.y, sv.z, sv.w};
        float te[4] = {tv.x, tv.y, tv.z, tv.w};
#pragma unroll
        for (int k = 0; k < 4; ++k)
            kd_accum(se[k], te[k], ms, mt, a0, a1, b0, b1, b2, b3, b4);
    }
    for (int j = (Cvec << 2) + tid; j < C; j += 256)   // tail (C % 4 = 1)
        kd_accum(s[j], t[j], ms, mt, a0, a1, b0, b1, b2, b3, b4);

    // ---- block max of ms, mt ----
    float wms = wave_max(ms);
    if (lane == 0) sMax[wave] = wms;
    __syncthreads();
    float Ms = -FLT_MAX;
#pragma unroll
    for (int i = 0; i < 8; ++i) Ms = fmaxf(Ms, sMax[i]);
    __syncthreads();
    float wmt = wave_max(mt);
    if (lane == 0) sMax[wave] = wmt;
    __syncthreads();
    float Mt = -FLT_MAX;
#pragma unroll
    for (int i = 0; i < 8; ++i) Mt = fmaxf(Mt, sMax[i]);

    // ---- rescale per-thread accumulators to the block maxima ----
    {
        float dsT = (ms - Ms) * INV_T;
        float edsT = __expf(dsT);
        float eds = edsT * edsT; eds *= eds;
        a0 *= eds; a1 *= edsT;

        float dt = mt - Mt;
        float edtT = __expf(dt * INV_T);
        float edt = edtT * edtT; edt *= edt;
        b1 = edt * (b1 + dt * b0);
        b0 *= edt; b2 *= edtT; b3 *= edtT; b4 *= edtT;
    }

    // ---- wave-level sums, partials into LDS (quantity q = "matrix row") ----
    float q[7] = {a0, a1, b0, b1, b2, b3, b4};
#pragma unroll
    for (int i = 0; i < 7; ++i) q[i] = wave_sum(q[i]);
    if (lane == 0) {
#pragma unroll
        for (int i = 0; i < 7; ++i) sP[wave][i] = q[i];
#pragma unroll
        for (int i = 7; i < 16; ++i) sP[wave][i] = 0.0f;
    }
    __syncthreads();

    // ---- cross-wave reduction with one V_WMMA_F32_16X16X4_F32 on wave 0 ----
    // A (16x4): row m = quantity m, A[m,k] = sum of wave-pair 2k..2k+1 partials.
    //   VGPR0: lanes0-15 -> K=0, lanes16-31 -> K=2 ; VGPR1: K=1 / K=3.
    // B (4x16) = ones  =>  D[m,n] = sum_k A[m,k] = block total of quantity m.
    // D layout: VGPR r, lanes 0-15 hold M=r  =>  lane 0, D[q] = total(q).
    if (wave == 0) {
        int m  = lane & 15;
        int hi = lane >> 4;                         // 0 -> K=0,1 ; 1 -> K=2,3
        v2f A;
        A[0] = sP[hi * 4 + 0][m] + sP[hi * 4 + 1][m];
        A[1] = sP[hi * 4 + 2][m] + sP[hi * 4 + 3][m];
        v2f Bv; Bv[0] = 1.0f; Bv[1] = 1.0f;
        v8f Cv = {};
        v8f D = __builtin_amdgcn_wmma_f32_16x16x4_f32(
            /*neg_a=*/false, A, /*neg_b=*/false, Bv,
            /*c_mod=*/(short)0, Cv, /*reuse_a=*/false, /*reuse_b=*/false);

        if (lane == 0) {
            float Zs  = D[0], ZsT = D[1], Zt = D[2], S1t = D[3];
            float ZtT = D[4], Wt  = D[5], Ws = D[6];

            float entropy = logf(Zt) - S1t / Zt;
            float alpha   = 1.0f - entropy / logf((float)C);
            alpha = fminf(fmaxf(alpha, 0.0f), 1.0f);

            float ce = Ms + logf(Zs) - s_tgt;

            float lse_sT = Ms * INV_T + logf(ZsT);
            float lse_tT = Mt * INV_T + logf(ZtT);
            float kl = (Wt - Ws) / (ZtT * T_DIST) + (lse_sT - lse_tT);

            row_loss[row] = (1.0f - alpha) * ce + alpha * (T_DIST * T_DIST) * kl;
        }
    }
}

__global__ __launch_bounds__(256)
void kd_mean_kernel(const float* __restrict__ row_loss, float* __restrict__ out, int B) {
    __shared__ float sW[8];
    int tid = threadIdx.x, lane = tid & 31, wave = tid >> 5;
    float v = 0.0f;
    for (int i = tid; i < B; i += 256) v += row_loss[i];
    v = wave_sum(v);
    if (lane == 0) sW[wave] = v;
    __syncthreads();
    if (tid == 0) {
        float s = 0.0f;
#pragma unroll
        for (int i = 0; i < 8; ++i) s += sW[i];
        out[0] = s / (float)B;
    }
}

extern "C" void kernel_launch(void* const* d_in, const int* in_sizes, int n_in,
                              void* d_out, int out_size, void* d_ws, size_t ws_size,
                              hipStream_t stream) {
    const float*     outputs = (const float*)d_in[0];
    const float*     teacher = (const float*)d_in[1];
    const long long* targets = (const long long*)d_in[2];   // int64 targets
    const int B = in_sizes[2];                              // 2048
    const int C = in_sizes[0] / B;                          // 50257

    float* row_loss = (float*)d_ws;                         // B floats of scratch

    kd_row_kernel<<<B, 256, 0, stream>>>(outputs, teacher, targets, row_loss, C);
    kd_mean_kernel<<<1, 256, 0, stream>>>(row_loss, (float*)d_out, B);
}